// EdgeConv2d_12841952215498
// MI455X (gfx1250) — compile-verified
//
#include <hip/hip_runtime.h>

// EdgeConv factorization:
//   W @ [x_i ; x_j - x_i] = (W1 - W2) @ x_i + W2 @ x_j
// Stage 1: per-node GEMM  y[b][m][0:64]  = (W1-W2) @ x[b,:,m] + bias
//                         y[b][m][64:128]=  W2     @ x[b,:,m]
//          via V_WMMA_F32_16X16X4_F32 (fp32 WMMA, shrinks 10.5 GFLOP edge
//          GEMM to 0.66 GFLOP node GEMM).
// Stage 2: per-edge gather relu(y1[i]+y2[j]) * 2*sigmoid(-dis), max over K.
//          y is 20 MB -> L2 resident (192 MB); gathers are 256 B contiguous.

#define Bn    2
#define Cn    64
#define Nn    20000
#define Kn    16
#define COUTn 64

typedef __attribute__((ext_vector_type(2))) float v2f;
typedef __attribute__((ext_vector_type(8))) float v8f;

// ---------------------------------------------------------------------------
// Wc[128][64]:  rows 0..63  = W[:, :64] - W[:, 64:]   (applies to x_i)
//               rows 64..127= W[:, 64:]               (applies to x_j)
// ---------------------------------------------------------------------------
__global__ void build_wc(const float* __restrict__ W, float* __restrict__ Wc) {
    int t = blockIdx.x * blockDim.x + threadIdx.x;
    if (t >= 128 * 64) return;
    int o = t >> 6, c = t & 63;
    float v;
    if (o < 64) v = W[o * 128 + c] - W[o * 128 + 64 + c];
    else        v = W[(o - 64) * 128 + 64 + c];
    Wc[t] = v;
}

// ---------------------------------------------------------------------------
// Node GEMM: one wave per 16 nodes. Output tile 128(o) x 16(m) via 8
// accumulators; K=64 in chunks of 4 -> 8*16 = 128 WMMA ops per wave.
// y layout: [B][N][128] (channel-contiguous -> 256B gathers in stage 2).
// ---------------------------------------------------------------------------
__global__ __launch_bounds__(32) void gemm_y(
    const float* __restrict__ x, const float* __restrict__ Wc,
    const float* __restrict__ bias, float* __restrict__ y) {
    const int lane = threadIdx.x;      // full wave32, EXEC all ones
    const int nlo  = lane & 15;
    const int hi   = lane >> 4;
    const int tile = blockIdx.x;                   // 0 .. B*(N/16)-1
    const int b    = tile / (Nn / 16);
    const int m0   = (tile % (Nn / 16)) * 16;

    // C/D layout: vgpr r holds (M = r + 8*hi, N = nlo). Fold bias into y1.
    v8f acc[8];
#pragma unroll
    for (int mt = 0; mt < 8; ++mt) {
#pragma unroll
        for (int r = 0; r < 8; ++r) {
            int o = mt * 16 + r + 8 * hi;
            acc[mt][r] = (o < COUTn) ? bias[o] : 0.0f;
        }
    }

    const float* xb = x + (size_t)b * Cn * Nn;
#pragma unroll 4
    for (int c0 = 0; c0 < Cn; c0 += 4) {
        // B fragment 4x16 (K x N): v0 = K{0|2}, v1 = K{1|3}, N = nlo
        v2f bf;
        bf.x = xb[(size_t)(c0 + 2 * hi)     * Nn + m0 + nlo];
        bf.y = xb[(size_t)(c0 + 2 * hi + 1) * Nn + m0 + nlo];
#pragma unroll
        for (int mt = 0; mt < 8; ++mt) {
            // A fragment 16x4 (M x K): M = nlo, v0 = K{0|2}, v1 = K{1|3}
            v2f af;
            af.x = Wc[(mt * 16 + nlo) * 64 + c0 + 2 * hi];
            af.y = Wc[(mt * 16 + nlo) * 64 + c0 + 2 * hi + 1];
            acc[mt] = __builtin_amdgcn_wmma_f32_16x16x4_f32(
                false, af, false, bf, (short)0, acc[mt], false, false);
        }
    }

    // Store: lane's column m = m0+nlo; acc[mt][0..7] are 8 consecutive
    // channels starting at mt*16 + 8*hi -> two b128 stores per tile.
    float* yrow = y + ((size_t)b * Nn + m0 + nlo) * 128;
#pragma unroll
    for (int mt = 0; mt < 8; ++mt) {
        float4* dst = (float4*)(yrow + mt * 16 + 8 * hi);
        dst[0] = make_float4(acc[mt][0], acc[mt][1], acc[mt][2], acc[mt][3]);
        dst[1] = make_float4(acc[mt][4], acc[mt][5], acc[mt][6], acc[mt][7]);
    }
}

// ---------------------------------------------------------------------------
// Edge stage: 1 wave per node (8 nodes / 256-thread block).
// Lanes 0..15 each own one neighbor k: load (i,j) indices + suppression,
// broadcast via wave32 shuffles. Each lane tracks channels {lane, lane+32}.
// Gathers of y columns are 256B contiguous, L2-resident.
// ---------------------------------------------------------------------------
__global__ __launch_bounds__(256) void edge_combine(
    const float* __restrict__ y, const int* __restrict__ eidx,
    const float* __restrict__ pos, float* __restrict__ out) {
    __shared__ float smem[64][8];
    const int t    = threadIdx.x;
    const int lane = t & 31;
    const int w    = t >> 5;                       // node slot in block
    const long p   = (long)blockIdx.x * 8 + w;     // B*N = 40000 = 5000*8 exact
    const int b    = (int)(p / Nn);
    const int n    = (int)(p - (long)b * Nn);

    int iv = 0, jv = 0;
    float supp = 0.0f;
    if (lane < Kn) {
        const size_t ebase = ((size_t)b * Nn + n) * Kn + lane;
        jv = eidx[ebase];                                // edge_index[0]: neighbor
        iv = eidx[(size_t)Bn * Nn * Kn + ebase];         // edge_index[1]: center
        const float* pb = pos + (size_t)b * 3 * Nn;
        float dx = pb[jv] - pb[iv];
        float dy = pb[Nn + jv] - pb[Nn + iv];
        float dz = pb[2 * Nn + jv] - pb[2 * Nn + iv];
        float dis = sqrtf(dx * dx + dy * dy + dz * dz);
        supp = 2.0f / (1.0f + __expf(dis));              // 2*sigmoid(-dis)
    }

    float mx0 = 0.0f, mx1 = 0.0f;                        // relu() >= 0
    const float* yb = y + (size_t)b * Nn * 128;
#pragma unroll
    for (int k = 0; k < Kn; ++k) {
        int   ik = __shfl(iv, k, 32);
        int   jk = __shfl(jv, k, 32);
        float sk = __shfl(supp, k, 32);
        const float* yi = yb + (size_t)ik * 128;         // y1 (+bias) at center
        const float* yj = yb + (size_t)jk * 128 + 64;    // y2 at neighbor
        float a0 = yi[lane]      + yj[lane];
        float a1 = yi[lane + 32] + yj[lane + 32];
        a0 = fmaxf(a0, 0.0f) * sk;
        a1 = fmaxf(a1, 0.0f) * sk;
        mx0 = fmaxf(mx0, a0);
        mx1 = fmaxf(mx1, a1);
    }

    // Stage through LDS so global stores are coalesced per channel row.
    smem[lane][w]      = mx0;
    smem[lane + 32][w] = mx1;
    __syncthreads();
    const int b0 = (blockIdx.x * 8) / Nn;               // same b for all 8 nodes
    const int n0 = blockIdx.x * 8 - b0 * Nn;
#pragma unroll
    for (int v = t; v < 512; v += 256) {
        int o = v >> 3, ni = v & 7;
        out[((size_t)b0 * COUTn + o) * Nn + n0 + ni] = smem[o][ni];
    }
}

extern "C" void kernel_launch(void* const* d_in, const int* in_sizes, int n_in,
                              void* d_out, int out_size, void* d_ws, size_t ws_size,
                              hipStream_t stream) {
    const float* x    = (const float*)d_in[0];   // [B, C, N, 1]
    const int*   eidx = (const int*)  d_in[1];   // [2, B, N, K]
    const float* pos  = (const float*)d_in[2];   // [B, 3, N, 1]
    const float* W    = (const float*)d_in[3];   // [COUT, 2C]
    const float* bias = (const float*)d_in[4];   // [COUT]

    float* Wc = (float*)d_ws;                                 // 128*64 floats (32 KB)
    float* y  = (float*)((char*)d_ws + 128 * 64 * sizeof(float)); // B*N*128 floats (20.5 MB)

    build_wc<<<(128 * 64 + 255) / 256, 256, 0, stream>>>(W, Wc);
    gemm_y<<<Bn * (Nn / 16), 32, 0, stream>>>(x, Wc, bias, y);
    edge_combine<<<(Bn * Nn) / 8, 256, 0, stream>>>(y, eidx, pos, (float*)d_out);
}